// RunningCenters_1881195675951
// MI455X (gfx1250) — compile-verified
//
#include <hip/hip_runtime.h>

#define DIMS 128
#define CHUNK 512

typedef __attribute__((ext_vector_type(2))) float v2f;
typedef __attribute__((ext_vector_type(8))) float v8f;

typedef __attribute__((address_space(1))) int* gint_p;
typedef __attribute__((address_space(3))) int* lint_p;

static __device__ __forceinline__ int imin(int a, int b) { return a < b ? a : b; }

// ---------------- pass 0: zero counters ----------------
__global__ void rc_zero(int* counts, int* cursors) {
    int i = threadIdx.x;            // one block of 1024
    counts[i]  = 0;
    cursors[i] = 0;
}

// ---------------- pass 1: label histogram ----------------
__global__ void rc_hist(const int* __restrict__ y, int* __restrict__ counts, int n) {
    int i = blockIdx.x * blockDim.x + threadIdx.x;
    int stride = gridDim.x * blockDim.x;
    for (; i < n; i += stride) {
        atomicAdd(&counts[y[i]], 1);
    }
}

// ---------------- pass 2: exclusive prefix scan (C <= 1024) ----------------
__global__ void rc_scan(const int* __restrict__ counts, int* __restrict__ offsets, int C) {
    __shared__ int s[1024];
    int t = threadIdx.x;
    int v = (t < C) ? counts[t] : 0;
    s[t] = v;
    __syncthreads();
    for (int d = 1; d < 1024; d <<= 1) {
        int add = (t >= d) ? s[t - d] : 0;
        __syncthreads();
        s[t] += add;
        __syncthreads();
    }
    if (t < C) offsets[t] = s[t] - v;   // exclusive prefix
}

// ---------------- pass 3: scatter point indices class-sorted ----------------
__global__ void rc_scatter(const int* __restrict__ y, const int* __restrict__ offsets,
                           int* __restrict__ cursors, int* __restrict__ order, int n) {
    int i = blockIdx.x * blockDim.x + threadIdx.x;
    int stride = gridDim.x * blockDim.x;
    for (; i < n; i += stride) {
        int c = y[i];
        int p = atomicAdd(&cursors[c], 1);
        order[offsets[c] + p] = i;
    }
}

// ---------------- pass 4: per-class gather-reduce via WMMA + CMA ----------------
// Grid = C blocks, 128 threads (4 wave32). Wave w owns dims [32w, 32w+32):
// two 16-wide column tiles with separate v8f accumulators. Row sums are
// D += ones(16x4) x B(4x16) on v_wmma_f32_16x16x4_f32; every row of D holds
// the running column sum, so acc[0] yields sum-over-rows per lane.
// Class row indices are staged into LDS via global_load_async_to_lds_b32
// (ASYNCcnt path) so the steady-state loop has unconditional, clause-able
// gather loads and no EXEC manipulation anywhere near the WMMAs.
__global__ void __launch_bounds__(128)
rc_reduce(const float* __restrict__ x, const float* __restrict__ centers,
          const float* __restrict__ counterp, const int* __restrict__ counts,
          const int* __restrict__ offsets, const int* __restrict__ order,
          float* __restrict__ out) {
    __shared__ int sidx[CHUNK];

    const int c    = blockIdx.x;
    const int lane = threadIdx.x & 31;
    const int wave = threadIdx.x >> 5;
    const int cnt  = counts[c];
    const int off  = offsets[c];
    const float ctr = counterp[0];

    const int half = lane >> 4;       // which B-VGPR row pair this lane feeds
    const int n    = lane & 15;       // column within 16-wide tile
    const int cb0  = wave * 32;       // tile 0 base dim
    const int cb1  = cb0 + 16;        // tile 1 base dim

    v2f ones; ones.x = 1.0f; ones.y = 1.0f;
    v8f acc0 = {};
    v8f acc1 = {};

    for (int base = 0; base < cnt; base += CHUNK) {
        const int m = imin(CHUNK, cnt - base);

        // ---- stage this chunk's indices into LDS (async copy) ----
        for (int t = threadIdx.x; t < m; t += 128) {
#if __has_builtin(__builtin_amdgcn_global_load_async_to_lds_b32)
            __builtin_amdgcn_global_load_async_to_lds_b32(
                (gint_p)(order + off + base + t), (lint_p)&sidx[t], 0, 0);
#else
            sidx[t] = order[off + base + t];
#endif
        }
#if __has_builtin(__builtin_amdgcn_global_load_async_to_lds_b32)
#if __has_builtin(__builtin_amdgcn_s_wait_asynccnt)
        __builtin_amdgcn_s_wait_asynccnt(0);
#else
        asm volatile("s_wait_asynccnt 0x0" ::: "memory");
#endif
#endif
        __syncthreads();

        // ---- main loop: groups of 8 rows, unconditional loads, 4 WMMAs ----
        const int full = m & ~7;
        for (int r = 0; r < full; r += 8) {
            const int ia0 = sidx[r + half];
            const int ia1 = sidx[r + 2 + half];
            const int ia2 = sidx[r + 4 + half];
            const int ia3 = sidx[r + 6 + half];
            const float* p0 = x + (long)ia0 * DIMS + n;
            const float* p1 = x + (long)ia1 * DIMS + n;
            const float* p2 = x + (long)ia2 * DIMS + n;
            const float* p3 = x + (long)ia3 * DIMS + n;
            v2f b00, b01, b10, b11;
            b00.x = p0[cb0]; b00.y = p1[cb0];
            b01.x = p0[cb1]; b01.y = p1[cb1];
            b10.x = p2[cb0]; b10.y = p3[cb0];
            b11.x = p2[cb1]; b11.y = p3[cb1];

            // prefetch the gather 16 rows ahead (clamped -> always valid, no branch)
            const int ip = sidx[imin(r + 16 + half, m - 1)];
            __builtin_prefetch((const void*)(x + (long)ip * DIMS + cb0 + n), 0, 0);

            acc0 = __builtin_amdgcn_wmma_f32_16x16x4_f32(
                false, ones, false, b00, (short)0, acc0, false, false);
            acc1 = __builtin_amdgcn_wmma_f32_16x16x4_f32(
                false, ones, false, b01, (short)0, acc1, false, false);
            acc0 = __builtin_amdgcn_wmma_f32_16x16x4_f32(
                false, ones, false, b10, (short)0, acc0, false, false);
            acc1 = __builtin_amdgcn_wmma_f32_16x16x4_f32(
                false, ones, false, b11, (short)0, acc1, false, false);
        }

        // ---- tail: clamped indices (valid addresses) + value-side selects ----
        for (int r = full; r < m; r += 4) {
            const int k0 = r + half;
            const int k1 = r + 2 + half;
            const bool v0 = k0 < m;
            const bool v1 = k1 < m;
            const int i0 = sidx[v0 ? k0 : m - 1];
            const int i1 = sidx[v1 ? k1 : m - 1];
            const float* q0 = x + (long)i0 * DIMS + n;
            const float* q1 = x + (long)i1 * DIMS + n;
            const float f00 = q0[cb0], f01 = q0[cb1];
            const float f10 = q1[cb0], f11 = q1[cb1];
            v2f b0, b1;
            b0.x = v0 ? f00 : 0.0f; b0.y = v1 ? f10 : 0.0f;
            b1.x = v0 ? f01 : 0.0f; b1.y = v1 ? f11 : 0.0f;
            acc0 = __builtin_amdgcn_wmma_f32_16x16x4_f32(
                false, ones, false, b0, (short)0, acc0, false, false);
            acc1 = __builtin_amdgcn_wmma_f32_16x16x4_f32(
                false, ones, false, b1, (short)0, acc1, false, false);
        }
        __syncthreads();   // protect sidx before next chunk overwrites it
    }

    // acc[0]: lanes 0..15 hold (M=0, N=0..15) = column sums for each tile
    const float s0 = acc0[0];
    const float s1 = acc1[0];
    if (lane < 16) {
        const int d0 = cb0 + n;
        const int d1 = cb1 + n;
        const float cen0 = centers[c * DIMS + d0];
        const float cen1 = centers[c * DIMS + d1];
        float r0, r1;
        if (cnt > 0) {
            const float inv = 1.0f / fmaxf((float)cnt, 1.0f);
            const float rcp1 = 1.0f / (ctr + 1.0f);
            r0 = (s0 * inv + cen0 * ctr) * rcp1;
            r1 = (s1 * inv + cen1 * ctr) * rcp1;
        } else {
            r0 = cen0;
            r1 = cen1;
        }
        out[c * DIMS + d0] = r0;
        out[c * DIMS + d1] = r1;
    }
}

extern "C" void kernel_launch(void* const* d_in, const int* in_sizes, int n_in,
                              void* d_out, int out_size, void* d_ws, size_t ws_size,
                              hipStream_t stream) {
    const float* x       = (const float*)d_in[0];
    const int*   y       = (const int*)d_in[1];
    const float* centers = (const float*)d_in[2];
    const float* counter = (const float*)d_in[3];
    float*       out     = (float*)d_out;

    const int N = in_sizes[1];          // number of points
    const int C = in_sizes[2] / DIMS;   // number of centers (reference: 1000)

    // workspace layout: counts[1024] | offsets[1024] | cursors[1024] | order[N]
    int* counts  = (int*)d_ws;
    int* offsets = counts + 1024;
    int* cursors = offsets + 1024;
    int* order   = cursors + 1024;

    int blocks = (N + 255) / 256;
    if (blocks > 4096) blocks = 4096;

    rc_zero   <<<1, 1024, 0, stream>>>(counts, cursors);
    rc_hist   <<<blocks, 256, 0, stream>>>(y, counts, N);
    rc_scan   <<<1, 1024, 0, stream>>>(counts, offsets, C);
    rc_scatter<<<blocks, 256, 0, stream>>>(y, offsets, cursors, order, N);
    rc_reduce <<<C, 128, 0, stream>>>(x, centers, counter, counts, offsets, order, out);
}